// MOECascadeModel_54606214202235
// MI455X (gfx1250) — compile-verified
//
#include <hip/hip_runtime.h>
#include <hip/hip_bf16.h>

// Problem constants (from reference): T=2048 tokens, H=7168 hidden, K=8 topk, E=128 experts.
#define T_TOKENS 2048
#define H_DIM    7168
#define K_TOP    8
#define H_VEC    (H_DIM / 4)        // 1792 float4 per row
#define BLK      256
#define VEC_PER_THREAD (H_VEC / BLK) // 7

typedef __attribute__((ext_vector_type(2))) float v2f;
typedef __attribute__((ext_vector_type(8))) float v8f;

// ---------------------------------------------------------------------------
// Kernel 1: per-token scale sums via V_WMMA_F32_16X16X4_F32.
// One wave32 computes sums for 16 tokens:
//   D = A(16x4 scales) x B(4x16 ones) + C, twice (K = 0..3 then 4..7).
// A-matrix 16x4 f32 VGPR layout (ISA 7.12.2): lanes 0-15 = M, VGPR0->K=0,
// VGPR1->K=1; lanes 16-31: VGPR0->K=2, VGPR1->K=3. B = all ones is
// layout-invariant. D row m holds the token's scale-sum in every column.
// C/D layout: VGPR j: lanes 0-15 -> M=j, lanes 16-31 -> M=8+j, N = lane&15.
// ---------------------------------------------------------------------------
__global__ __launch_bounds__(32) void moe_scale_sum_wmma(
    const float* __restrict__ scales, float* __restrict__ sums) {
  const int lane = threadIdx.x;            // 0..31, full wave active (EXEC all 1s)
  const int base = blockIdx.x * 16;        // 16 tokens per wave
  const int m    = lane & 15;              // token row within tile
  const int kg   = (lane >> 4) * 2;        // K group: 0 (lanes 0-15) or 2 (16-31)

  const float* row = scales + (size_t)(base + m) * K_TOP;

  v2f a0, a1, b;
  a0.x = row[kg + 0];
  a0.y = row[kg + 1];
  a1.x = row[kg + 4];
  a1.y = row[kg + 5];
  b.x = 1.0f;
  b.y = 1.0f;

  v8f c = {};
  // 8 args: (neg_a, A, neg_b, B, c_mod, C, reuse_a, reuse_b)
  c = __builtin_amdgcn_wmma_f32_16x16x4_f32(false, a0, false, b, (short)0, c,
                                            false, false);
  c = __builtin_amdgcn_wmma_f32_16x16x4_f32(false, a1, false, b, (short)0, c,
                                            false, false);

  // Lanes with N==0 (lane 0 and lane 16) hold M = j and M = 8+j in c[j].
  if (m == 0) {
    const int moff = (lane >> 4) * 8;
#pragma unroll
    for (int j = 0; j < 8; ++j) {
      sums[base + moff + j] = c[j];
    }
  }
}

// ---------------------------------------------------------------------------
// Kernel 2: streaming combine. out[t,:] = mask[t] ? sum[t]*x[t,:] : ori_x[t,:]
// One block per token row; uniform scalar branch on mask so each row reads
// exactly one 28.7KB input row (single-pass, HBM/L2 bandwidth bound).
// ---------------------------------------------------------------------------
__global__ __launch_bounds__(BLK) void moe_combine_stream(
    const float4* __restrict__ x, const float4* __restrict__ ori_x,
    const float* __restrict__ sums, const unsigned char* __restrict__ mask,
    float4* __restrict__ out) {
  const int t = blockIdx.x;
  const size_t rowbase = (size_t)t * H_VEC;

  if (mask[t]) {                     // uniform per block -> scalar branch
    const float s = sums[t];
#pragma unroll
    for (int i = 0; i < VEC_PER_THREAD; ++i) {
      const size_t idx = rowbase + threadIdx.x + i * BLK;
      float4 v = x[idx];
      v.x *= s; v.y *= s; v.z *= s; v.w *= s;
      out[idx] = v;
    }
  } else {
#pragma unroll
    for (int i = 0; i < VEC_PER_THREAD; ++i) {
      const size_t idx = rowbase + threadIdx.x + i * BLK;
      out[idx] = ori_x[idx];
    }
  }
}

// ---------------------------------------------------------------------------
// Inputs (setup_inputs order):
//   d_in[0] x             float32 [T, H]
//   d_in[1] expert_scales float32 [T, K]
//   d_in[2] ori_x         float32 [T, H]
//   d_in[3] expert_ids    int32   [T, K]   (dead: permutation roundtrip = identity)
//   d_in[4] x_active_mask bool    [T]      (1 byte per element)
// Output: float32 [T, H]
// Workspace: first T floats = per-token scale sums.
// ---------------------------------------------------------------------------
extern "C" void kernel_launch(void* const* d_in, const int* in_sizes, int n_in,
                              void* d_out, int out_size, void* d_ws, size_t ws_size,
                              hipStream_t stream) {
  const float*         x      = (const float*)d_in[0];
  const float*         scales = (const float*)d_in[1];
  const float*         ori    = (const float*)d_in[2];
  const unsigned char* mask   = (const unsigned char*)d_in[4];
  float* out  = (float*)d_out;
  float* sums = (float*)d_ws;

  moe_scale_sum_wmma<<<T_TOKENS / 16, 32, 0, stream>>>(scales, sums);
  moe_combine_stream<<<T_TOKENS, BLK, 0, stream>>>(
      (const float4*)x, (const float4*)ori, sums, mask, (float4*)out);
}